// YOLOLoss_35931696398924
// MI455X (gfx1250) — compile-verified
//
#include <hip/hip_runtime.h>
#include <math.h>

typedef __attribute__((ext_vector_type(2))) float v2f;
typedef __attribute__((ext_vector_type(8))) float v8f;

#define CPB 256                 // cells per block
#define FPC 30                  // floats per cell (5*B + C = 30)
#define TILE_F (CPB * FPC)      // 7680 floats per tile
#define NCLS 20

// Sum v across all 32 lanes of the wave using V_WMMA_F32_16X16X4_F32.
// A (16x4) per-lane layout: lane m (0..15) holds A[m][0]=VGPR0, A[m][1]=VGPR1;
// lane m+16 holds A[m][2],A[m][3].  With A=(v,0) and B=ones(4x16):
//   D[m][n] = v[m] + v[m+16]  (independent of n)
// D VGPR r at lane l = s[r + (l>=16 ? 8 : 0)], so sum(d[0..7]) gives the
// low-half sum in lanes 0-15 and the high-half sum in lanes 16-31; one
// xor-16 shuffle completes the 32-lane fp32-exact reduction.
__device__ __forceinline__ float wave_sum32(float v) {
  v2f a; a.x = v;    a.y = 0.0f;
  v2f b; b.x = 1.0f; b.y = 1.0f;
  v8f c = {};
  v8f d = __builtin_amdgcn_wmma_f32_16x16x4_f32(
      /*neg_a=*/false, a, /*neg_b=*/false, b,
      /*c_mod=*/(short)0, c, /*reuse_a=*/false, /*reuse_b=*/false);
  float t = d[0] + d[1] + d[2] + d[3] + d[4] + d[5] + d[6] + d[7];
  t += __shfl_xor(t, 16, 32);
  return t;
}

__global__ void init_ws_kernel(float* acc) {
  if (threadIdx.x < 8) acc[threadIdx.x] = 0.0f;
}

__global__ __launch_bounds__(CPB) void yolo_main_kernel(
    const float* __restrict__ pred, const float* __restrict__ target,
    const int* __restrict__ choice, float* __restrict__ acc, int n_cells) {
  __shared__ float sp[TILE_F];
  __shared__ float st[TILE_F];
  __shared__ float wred[8 * 6];

  const int tid = threadIdx.x;
  const long long base_cell = (long long)blockIdx.x * CPB;
  const int cells = min(CPB, n_cells - (int)base_cell);
  const long long base_f = base_cell * FPC;
  const int nf = cells * FPC;

  // Stage tiles into LDS with coalesced b128 loads (tile base is 16B aligned:
  // blockIdx * 7680 floats = blockIdx * 30720 bytes).
  if (nf == TILE_F) {
    const float4* gp = (const float4*)(pred + base_f);
    const float4* gt = (const float4*)(target + base_f);
    float4* lp = (float4*)sp;
    float4* lt = (float4*)st;
    for (int i = tid; i < TILE_F / 4; i += CPB) {
      lp[i] = gp[i];
      lt[i] = gt[i];
    }
  } else {
    for (int i = tid; i < nf; i += CPB) {
      sp[i] = pred[base_f + i];
      st[i] = target[base_f + i];
    }
  }
  __syncthreads();

  const bool valid = (tid < cells);
  const int ci = valid ? tid : 0;            // tail lanes read cell 0, masked later
  const float* pd = sp + ci * FPC;
  const float* tg = st + ci * FPC;
  const int ch = choice[base_cell + ci];     // coalesced 4B loads
  const bool c = (ch != 0);

  const float tconf = tg[4];
  const float m  = (tconf > 0.0f) ? 1.0f : 0.0f;
  const float nm = (tconf == 0.0f) ? 1.0f : 0.0f;

  const int o = c ? 0 : 5;
  const float px = pd[o + 0], py = pd[o + 1], pw = pd[o + 2], ph = pd[o + 3];
  const float pobj = pd[o + 4];
  const float tx = tg[o + 0], ty = tg[o + 1];
  float tw = tg[o + 2], th = tg[o + 3];
  if (!(tconf > 0.0f)) { tw = 1.0f; th = 1.0f; }
  const float tobj = tg[o + 4];               // c -> tconf, !c -> target[9]

  const float dx = px - tx, dy = py - ty;
  const float dw = pw - sqrtf(tw), dh = ph - sqrtf(th);
  float loc   = m * (0.5f * (dx * dx + dy * dy) + 0.5f * (dw * dw + dh * dh));
  float objl  = m * (pobj - tobj) * (pobj - tobj);
  const float d4 = pd[4] - tconf, d9 = pd[9] - tg[9];
  float noobj = nm * (d4 * d4 + d9 * d9);

  // CE term: reference computes logp = log_softmax(softmax(logits)),
  // i.e. logp_j = sm_j - log(sum_k exp(sm_k)).
  float lg[NCLS];
#pragma unroll
  for (int j = 0; j < NCLS; ++j) lg[j] = pd[10 + j];
  float mx = lg[0];
#pragma unroll
  for (int j = 1; j < NCLS; ++j) mx = fmaxf(mx, lg[j]);

  int tcls = 0;
  float bt = tg[10];
#pragma unroll
  for (int j = 1; j < NCLS; ++j) {
    float tv = tg[10 + j];
    if (tv > bt) { bt = tv; tcls = j; }       // strict > keeps first max (jnp.argmax)
  }

  float Z = 0.0f;
#pragma unroll
  for (int j = 0; j < NCLS; ++j) {
    float e = __expf(lg[j] - mx);
    lg[j] = e;
    Z += e;
  }
  float et = lg[0];
#pragma unroll
  for (int j = 1; j < NCLS; ++j) et = (j == tcls) ? lg[j] : et;
  const float invZ = 1.0f / Z;
  float Z2 = 0.0f;
#pragma unroll
  for (int j = 0; j < NCLS; ++j) Z2 += __expf(lg[j] * invZ);
  const float ce = __logf(Z2) - et * invZ;    // -(sm_t - log Z2)
  float cls = m * ce;

  const float vmask = valid ? 1.0f : 0.0f;
  loc *= vmask; objl *= vmask; noobj *= vmask; cls *= vmask;
  const float mm = m * vmask, nn = nm * vmask;

  // Wave-level fp32 reduction via WMMA (EXEC is all-ones here: no divergence).
  const float r0 = wave_sum32(loc);
  const float r1 = wave_sum32(objl);
  const float r2 = wave_sum32(noobj);
  const float r3 = wave_sum32(cls);
  const float r4 = wave_sum32(mm);
  const float r5 = wave_sum32(nn);

  const int wave = tid >> 5;
  const int lane = tid & 31;
  if (lane == 0) {
    wred[wave * 6 + 0] = r0;
    wred[wave * 6 + 1] = r1;
    wred[wave * 6 + 2] = r2;
    wred[wave * 6 + 3] = r3;
    wred[wave * 6 + 4] = r4;
    wred[wave * 6 + 5] = r5;
  }
  __syncthreads();
  if (tid == 0) {
    float s0 = 0.f, s1 = 0.f, s2 = 0.f, s3 = 0.f, s4 = 0.f, s5 = 0.f;
#pragma unroll
    for (int w = 0; w < 8; ++w) {
      s0 += wred[w * 6 + 0];
      s1 += wred[w * 6 + 1];
      s2 += wred[w * 6 + 2];
      s3 += wred[w * 6 + 3];
      s4 += wred[w * 6 + 4];
      s5 += wred[w * 6 + 5];
    }
    atomicAdd(&acc[0], s0);   // sum m * loc_cell
    atomicAdd(&acc[1], s1);   // sum m * (pobj - tobj)^2
    atomicAdd(&acc[2], s2);   // sum nm * ((p4-tc)^2 + (p9-t9)^2)
    atomicAdd(&acc[3], s3);   // sum m * ce
    atomicAdd(&acc[4], s4);   // n_obj count
    atomicAdd(&acc[5], s5);   // n_no count
  }
}

__global__ void finalize_kernel(const float* __restrict__ acc,
                                float* __restrict__ out) {
  if (threadIdx.x == 0 && blockIdx.x == 0) {
    const float n_obj = fmaxf(acc[4], 1.0f);
    const float n_no  = fmaxf(acc[5], 1.0f);
    out[0] = 5.0f * acc[0];       // loc_loss  (L_COORD = 5)
    out[1] = acc[3] / n_obj;      // cls_loss
    out[2] = acc[1];              // obj_loss
    out[3] = 0.5f * acc[2] / n_no;// noobj_loss (L_NOOBJ = 0.5, both terms / n_no)
  }
}

extern "C" void kernel_launch(void* const* d_in, const int* in_sizes, int n_in,
                              void* d_out, int out_size, void* d_ws, size_t ws_size,
                              hipStream_t stream) {
  const float* pred   = (const float*)d_in[0];
  const float* target = (const float*)d_in[1];
  const int*   choice = (const int*)d_in[2];
  float* out = (float*)d_out;
  float* acc = (float*)d_ws;

  const int n_cells = in_sizes[2];          // BATCH * S * S = 802816
  const int blocks = (n_cells + CPB - 1) / CPB;

  init_ws_kernel<<<1, 32, 0, stream>>>(acc);
  yolo_main_kernel<<<blocks, CPB, 0, stream>>>(pred, target, choice, acc, n_cells);
  finalize_kernel<<<1, 32, 0, stream>>>(acc, out);
}